// GNN_Model_38946763441101
// MI455X (gfx1250) — compile-verified
//
#include <hip/hip_runtime.h>

#define N_NODES 50000
#define N_EDGES 400000
#define F_IN 16
#define F_OUT 32
#define S_EDGE 4

typedef __attribute__((ext_vector_type(2))) float v2f;
typedef __attribute__((ext_vector_type(8))) float v8f;

// Workspace layout (floats):
//   ws[0..63]   M[s][f]  (s*16+f)   = sum_e edge_attr[e,s] * x[src[e],f]
//   ws[64..79]  g[f]                = sum_e x[src[e],f]
//   ws[80..95]  xsum[f]             = sum_n x[n,f]

__global__ void gnn_zero_ws(float* __restrict__ ws) {
    int i = threadIdx.x;
    if (i < 96) ws[i] = 0.0f;
}

__global__ __launch_bounds__(256) void gnn_edge_accum(
    const float* __restrict__ x,
    const int*   __restrict__ edge_index,
    const float* __restrict__ edge_attr,
    float*       __restrict__ ws)
{
    __shared__ float smem[96];
    for (int i = threadIdx.x; i < 96; i += blockDim.x) smem[i] = 0.0f;
    __syncthreads();

    const int tid    = blockIdx.x * blockDim.x + threadIdx.x;
    const int lane   = threadIdx.x & 31;          // wave32
    const int m      = lane & 15;                 // A row (=s idx) / B col (=f idx)
    const int hi     = lane >> 4;                 // half-wave: K pair selector
    const int waveId = tid >> 5;
    const int nWaves = (gridDim.x * blockDim.x) >> 5;
    const int nPairs = N_EDGES / 8;               // 8 edges (2 WMMAs) per trip
    const int* __restrict__ src = edge_index + N_EDGES;  // row 1 of edge_index

    // Per-lane constants for the A matrix fill (branchless select)
    const bool  isAttr = (m < S_EDGE);
    const float fill   = (m == S_EDGE) ? 1.0f : 0.0f;  // row 4 accumulates g[f]
    const int   mc     = isAttr ? m : (S_EDGE - 1);    // clamped attr column

    v8f c = {0.f, 0.f, 0.f, 0.f, 0.f, 0.f, 0.f, 0.f};

    for (int p = waveId; p < nPairs; p += nWaves) {
        const int e  = p * 8;
        // A 16x4 layout: lanes 0-15 hold K=0 (vgpr0), K=1 (vgpr1);
        //                lanes 16-31 hold K=2, K=3.  Mirror for B rows.
        const int k0 = e + 2 * hi;                // edges for WMMA #0
        const int k1 = k0 + 4;                    // edges for WMMA #1

        // Prefetch next trip's index / attr cachelines (speculative)
        __builtin_prefetch(src + k0 + nWaves * 8, 0, 1);
        __builtin_prefetch(edge_attr + (size_t)(e + nWaves * 8) * S_EDGE, 0, 1);

        // ---- issue all independent loads up front ----
        const int2 i0 = *(const int2*)(src + k0); // src[k0], src[k0+1]
        const int2 i1 = *(const int2*)(src + k1); // src[k1], src[k1+1]

        const float a0r = edge_attr[(k0    ) * S_EDGE + mc];
        const float a1r = edge_attr[(k0 + 1) * S_EDGE + mc];
        const float a2r = edge_attr[(k1    ) * S_EDGE + mc];
        const float a3r = edge_attr[(k1 + 1) * S_EDGE + mc];

        v2f b0, b1;
        b0.x = x[i0.x * F_IN + m];
        b0.y = x[i0.y * F_IN + m];
        b1.x = x[i1.x * F_IN + m];
        b1.y = x[i1.y * F_IN + m];

        v2f a0, a1;
        a0.x = isAttr ? a0r : fill;
        a0.y = isAttr ? a1r : fill;
        a1.x = isAttr ? a2r : fill;
        a1.y = isAttr ? a3r : fill;

        // D = A(16x4) * B(4x16) + C ; rows 0..3 -> M[s,f], row 4 -> g[f]
        c = __builtin_amdgcn_wmma_f32_16x16x4_f32(
                false, a0, false, b0, (short)0, c, false, false);
        c = __builtin_amdgcn_wmma_f32_16x16x4_f32(
                false, a1, false, b1, (short)0, c, false, false);
    }

    // C/D layout: vgpr r, lanes 0-15 = row r, lanes 16-31 = row r+8.
    // Rows 0..4 live in lanes 0..15, c[0..4].
    if (hi == 0) {
        atomicAdd(&smem[ 0 + m], c[0]);
        atomicAdd(&smem[16 + m], c[1]);
        atomicAdd(&smem[32 + m], c[2]);
        atomicAdd(&smem[48 + m], c[3]);
        atomicAdd(&smem[64 + m], c[4]);
    }

    // xsum[f] = sum_n x[n,f] via float4 streaming; total threads T is a
    // multiple of 4 so the 4-column window per thread is loop-invariant.
    const int T   = gridDim.x * blockDim.x;
    const int nV4 = (N_NODES * F_IN) / 4;         // 200000 float4s
    const float4* __restrict__ x4 = (const float4*)x;
    float4 s4 = make_float4(0.f, 0.f, 0.f, 0.f);
    for (int i = tid; i < nV4; i += T) {
        float4 v = x4[i];
        s4.x += v.x; s4.y += v.y; s4.z += v.z; s4.w += v.w;
    }
    const int col = (tid * 4) & 15;
    atomicAdd(&smem[80 + col + 0], s4.x);
    atomicAdd(&smem[80 + col + 1], s4.y);
    atomicAdd(&smem[80 + col + 2], s4.z);
    atomicAdd(&smem[80 + col + 3], s4.w);

    __syncthreads();
    for (int i = threadIdx.x; i < 96; i += blockDim.x)
        atomicAdd(&ws[i], smem[i]);
}

__global__ void gnn_finalize(
    const float* __restrict__ ws,
    const float* __restrict__ W_kn,   const float* __restrict__ b_kn,
    const float* __restrict__ W_root, const float* __restrict__ b_conv,
    const float* __restrict__ W_dense,const float* __restrict__ b_dense,
    float* __restrict__ out)
{
    __shared__ float pooled[F_OUT];
    const int h = threadIdx.x;
    if (h < F_OUT) {
        float acc = (float)N_NODES * b_conv[h];
        #pragma unroll
        for (int f = 0; f < F_IN; ++f) {
            acc += ws[80 + f] * W_root[f * F_OUT + h];   // (sum_n x) @ W_root
            acc += ws[64 + f] * b_kn[f * F_OUT + h];     // g @ b_kn part
            #pragma unroll
            for (int s = 0; s < S_EDGE; ++s)
                acc += ws[s * 16 + f] * W_kn[s * (F_IN * F_OUT) + f * F_OUT + h];
        }
        pooled[h] = acc;
    }
    __syncthreads();
    if (h < 3) {
        float o = b_dense[h];
        #pragma unroll
        for (int j = 0; j < F_OUT; ++j) o += pooled[j] * W_dense[j * 3 + h];
        out[h] = fmaxf(o, 0.0f);
    }
}

extern "C" void kernel_launch(void* const* d_in, const int* in_sizes, int n_in,
                              void* d_out, int out_size, void* d_ws, size_t ws_size,
                              hipStream_t stream) {
    const float* x          = (const float*)d_in[0];
    const int*   edge_index = (const int*)  d_in[1];
    const float* edge_attr  = (const float*)d_in[2];
    const float* W_kn       = (const float*)d_in[3];
    const float* b_kn       = (const float*)d_in[4];
    const float* W_root     = (const float*)d_in[5];
    const float* b_conv     = (const float*)d_in[6];
    const float* W_dense    = (const float*)d_in[7];
    const float* b_dense    = (const float*)d_in[8];
    float*       out        = (float*)d_out;
    float*       ws         = (float*)d_ws;

    gnn_zero_ws<<<1, 128, 0, stream>>>(ws);
    gnn_edge_accum<<<240, 256, 0, stream>>>(x, edge_index, edge_attr, ws);
    gnn_finalize<<<1, 32, 0, stream>>>(ws, W_kn, b_kn, W_root, b_conv,
                                       W_dense, b_dense, out);
}